// UpdateEdge_21062519620180
// MI455X (gfx1250) — compile-verified
//
#include <hip/hip_runtime.h>
#include <cstddef>

// ---------------------------------------------------------------------------
// MI455X (gfx1250) implementation.
// All heavy math (MoE GEMMs, We, T-projection, both MLPs) runs on the matrix
// pipe via V_WMMA_F32_16X16X4_F32 (exact fp32 -> safe vs fp32 reference).
// Edges are the M dimension; each wave32 owns a 32x16 output block (two 16x16
// C tiles sharing one B fragment) so every strided B load feeds two WMMAs.
// ---------------------------------------------------------------------------

#define E_CNT 65536

typedef float v2f __attribute__((ext_vector_type(2)));
typedef float v8f __attribute__((ext_vector_type(8)));

__device__ __forceinline__ float silu_f(float x) { return x / (1.f + expf(-x)); }
__device__ __forceinline__ float sigm_f(float x) { return 1.f / (1.f + expf(-x)); }

// ---------------------------------------------------------------------------
// Generic mixture GEMM:  out[m, n] = act( Σ_k gf[m,k] * (A[m,:] @ W[k])[n] + bias[n] )
// A row m comes from A1 (optionally gathered via a_idx) for cols < K1, then A2.
// W element [k][f][n] at W + k*wk_stride + f*w_rs + n*w_cs (supports B^T).
// One wave per 32(M) x 16(N) block: two C tiles, shared B fragment.
// Fragment layout per ISA 7.12.2: lanes 0-15 hold K-pair {0,1}, lanes 16-31 {2,3};
// C: VGPR r -> row M = r + 8*(lane>=16), col N = lane&15.
// ---------------------------------------------------------------------------
__global__ __launch_bounds__(32) void moe_gemm_kernel(
    const float* __restrict__ A1, int lda1, int K1,
    const float* __restrict__ A2, int lda2, int K2,
    const int* __restrict__ a_idx,
    const float* __restrict__ W, int wk_stride, int w_rs, int w_cs,
    int NK, const float* __restrict__ gf,
    const float* __restrict__ bias, int act,
    float* __restrict__ out, int out_stride)
{
    const int m0    = blockIdx.x * 32;
    const int n0    = blockIdx.y * 16;
    const int lane  = threadIdx.x;
    const int mlo   = lane & 15;
    const int koff  = (lane >> 4) << 1;         // 0 for lanes 0-15, 2 for 16-31
    const int ncol  = n0 + mlo;
    const int mrow0 = m0 + mlo;
    const int mrow1 = mrow0 + 16;
    const int arow0 = a_idx ? a_idx[mrow0] : mrow0;
    const int arow1 = a_idx ? a_idx[mrow1] : mrow1;
    const int mb0   = m0 + ((lane >> 4) << 3);  // C row base, tile 0
    const int mb1   = mb0 + 16;                 // C row base, tile 1

    v8f acc0 = {}, acc1 = {};
    for (int k = 0; k < NK; ++k) {
        v8f c0 = {}, c1 = {};
        const float* wk  = W + (size_t)k * (size_t)wk_stride;
        // ---- A1 segment (uniform, no per-step predicate) ----
        {
            const float* a0p = A1 + (size_t)arow0 * lda1 + koff;
            const float* a1p = A1 + (size_t)arow1 * lda1 + koff;
            for (int kk = 0; kk < K1; kk += 4) {
                v2f a0, a1, b;
                a0.x = a0p[kk];  a0.y = a0p[kk + 1];
                a1.x = a1p[kk];  a1.y = a1p[kk + 1];
                const float* bp = wk + (size_t)(kk + koff) * w_rs + (size_t)ncol * w_cs;
                b.x = bp[0];
                b.y = bp[w_rs];
                c0 = __builtin_amdgcn_wmma_f32_16x16x4_f32(false, a0, false, b,
                                                           (short)0, c0, false, false);
                c1 = __builtin_amdgcn_wmma_f32_16x16x4_f32(false, a1, false, b,
                                                           (short)0, c1, false, false);
            }
        }
        // ---- optional A2 segment (concat input, direct-indexed) ----
        if (K2 > 0) {
            const float* a0p = A2 + (size_t)mrow0 * lda2 + koff;
            const float* a1p = A2 + (size_t)mrow1 * lda2 + koff;
            const float* wk2 = wk + (size_t)K1 * w_rs;
            for (int kk = 0; kk < K2; kk += 4) {
                v2f a0, a1, b;
                a0.x = a0p[kk];  a0.y = a0p[kk + 1];
                a1.x = a1p[kk];  a1.y = a1p[kk + 1];
                const float* bp = wk2 + (size_t)(kk + koff) * w_rs + (size_t)ncol * w_cs;
                b.x = bp[0];
                b.y = bp[w_rs];
                c0 = __builtin_amdgcn_wmma_f32_16x16x4_f32(false, a0, false, b,
                                                           (short)0, c0, false, false);
                c1 = __builtin_amdgcn_wmma_f32_16x16x4_f32(false, a1, false, b,
                                                           (short)0, c1, false, false);
            }
        }
        if (gf) {
#pragma unroll
            for (int r = 0; r < 8; ++r) {
                acc0[r] += gf[(size_t)(mb0 + r) * 16 + k] * c0[r];
                acc1[r] += gf[(size_t)(mb1 + r) * 16 + k] * c1[r];
            }
        } else {
            acc0 = c0;
            acc1 = c1;
        }
    }

    const float bj = bias ? bias[ncol] : 0.f;
#pragma unroll
    for (int r = 0; r < 8; ++r) {
        float v0 = acc0[r] + bj;
        float v1 = acc1[r] + bj;
        if (act) {
            v0 = v0 / (1.f + expf(-v0));
            v1 = v1 / (1.f + expf(-v1));
        }
        out[(size_t)(mb0 + r) * out_stride + ncol] = v0;
        out[(size_t)(mb1 + r) * out_stride + ncol] = v1;
    }
}

// ---------------------------------------------------------------------------
// Prep: gathers, Wigner rotations y1/y2, builds F0 (336), F1 (fp1|fm1, 288),
// F2 (fp2|fm2, 96), gf softmax (padded to 16), LayerNorm -> CAT1[:,0:128],
// and copies wigner_D1/D2 to the output. One block (128 thr) per edge.
// ---------------------------------------------------------------------------
__global__ __launch_bounds__(128) void prep_kernel(
    const float* __restrict__ latents, const float* __restrict__ nodef,
    const float* __restrict__ edgef,  const float* __restrict__ wD1,
    const float* __restrict__ wD2,    const float* __restrict__ moleg,
    const float* __restrict__ Wg,     const float* __restrict__ ln_g,
    const float* __restrict__ ln_b,   const int* __restrict__ eidx,
    const int* __restrict__ aedges,
    float* __restrict__ F0, float* __restrict__ F1, float* __restrict__ F2,
    float* __restrict__ GF, float* __restrict__ CAT1,
    float* __restrict__ outD1, float* __restrict__ outD2)
{
    const int e = blockIdx.x;
    const int t = threadIdx.x;
    const int ae = aedges[e];
    const int ec = eidx[ae];
    const int en = eidx[E_CNT + ae];

    __shared__ float sD1[9], sD2[25], sg[8], red[128];

    if (t < 9)  { float v = wD1[(size_t)e * 9  + t]; sD1[t] = v; outD1[(size_t)e * 9  + t] = v; }
    if (t < 25) { float v = wD2[(size_t)e * 25 + t]; sD2[t] = v; outD2[(size_t)e * 25 + t] = v; }

    if (t < 64) {  // x0 = [nc0, ef0, nn0]
        F0[(size_t)e * 336 + t]       = nodef[(size_t)ec * 240 + t];
        F0[(size_t)e * 336 + 64 + t]  = edgef[(size_t)e  * 240 + t];
        F0[(size_t)e * 336 + 128 + t] = nodef[(size_t)en * 240 + t];
    }
    __syncthreads();

    if (t < 96) {  // y1 = D1 @ x1 per channel
        const float* src;
        if (t < 32)      src = nodef + (size_t)ec * 240 + 64 + t * 3;
        else if (t < 64) src = edgef + (size_t)e  * 240 + 64 + (t - 32) * 3;
        else             src = nodef + (size_t)en * 240 + 64 + (t - 64) * 3;
        const float x0 = src[0], x1 = src[1], x2 = src[2];
        const float y0 = sD1[0]*x0 + sD1[1]*x1 + sD1[2]*x2;
        const float y1 = sD1[3]*x0 + sD1[4]*x1 + sD1[5]*x2;
        const float y2 = sD1[6]*x0 + sD1[7]*x1 + sD1[8]*x2;
        F0[(size_t)e * 336 + 192 + t] = y1;   // f0  <- y1[:,1]
        F1[(size_t)e * 288 + t]       = y2;   // fp1 <- y1[:,2]
        F1[(size_t)e * 288 + 144 + t] = y0;   // fm1 <- y1[:,0]
    }
    if (t < 48) {  // y2 = D2 @ x2 per channel
        const float* src;
        if (t < 16)      src = nodef + (size_t)ec * 240 + 160 + t * 5;
        else if (t < 32) src = edgef + (size_t)e  * 240 + 160 + (t - 16) * 5;
        else             src = nodef + (size_t)en * 240 + 160 + (t - 32) * 5;
        float x[5], y[5];
#pragma unroll
        for (int j = 0; j < 5; ++j) x[j] = src[j];
#pragma unroll
        for (int i = 0; i < 5; ++i) {
            float s = 0.f;
#pragma unroll
            for (int j = 0; j < 5; ++j) s += sD2[i * 5 + j] * x[j];
            y[i] = s;
        }
        F0[(size_t)e * 336 + 288 + t]      = y[2];  // f0  <- y2[:,2]
        F1[(size_t)e * 288 + 96 + t]       = y[3];  // fp1 <- y2[:,3]
        F1[(size_t)e * 288 + 240 + t]      = y[1];  // fm1 <- y2[:,1]
        F2[(size_t)e * 96 + t]             = y[4];  // fp2 <- y2[:,4]
        F2[(size_t)e * 96 + 48 + t]        = y[0];  // fm2 <- y2[:,0]
    }
    if (t < 8) {   // gating logits
        float s = 0.f;
        for (int f = 0; f < 64; ++f) s += moleg[(size_t)e * 64 + f] * Wg[f * 8 + t];
        sg[t] = s;
    }
    __syncthreads();
    if (t == 0) {  // softmax + append 1
        float mx = sg[0];
        for (int k = 1; k < 8; ++k) mx = fmaxf(mx, sg[k]);
        float ex[8], sum = 0.f;
        for (int k = 0; k < 8; ++k) { ex[k] = expf(sg[k] - mx); sum += ex[k]; }
        const float inv = 1.f / sum;
        for (int k = 0; k < 8; ++k) GF[(size_t)e * 16 + k] = ex[k] * inv;
        GF[(size_t)e * 16 + 8] = 1.f;
        for (int k = 9; k < 16; ++k) GF[(size_t)e * 16 + k] = 0.f;
    }

    // LayerNorm of latents[ae] -> CAT1[:,0:128]
    const float lv = latents[(size_t)ae * 128 + t];
    red[t] = lv;
    __syncthreads();
    for (int s = 64; s > 0; s >>= 1) { if (t < s) red[t] += red[t + s]; __syncthreads(); }
    const float mu = red[0] * (1.f / 128.f);
    __syncthreads();
    const float dv = lv - mu;
    red[t] = dv * dv;
    __syncthreads();
    for (int s = 64; s > 0; s >>= 1) { if (t < s) red[t] += red[t + s]; __syncthreads(); }
    const float var = red[0] * (1.f / 128.f);
    CAT1[(size_t)e * 192 + t] = dv * rsqrtf(var + 1e-5f) * ln_g[t] + ln_b[t];
}

// ---------------------------------------------------------------------------
// Middle: combine complex MoE parts, back-rotate with D^T, gates, P0/P1/P2
// channel mixes, We scaling -> NEWE (240); scalars -> CAT1[:,128:192].
// One block (64 thr) per edge.
// ---------------------------------------------------------------------------
__global__ __launch_bounds__(64) void middle_kernel(
    const float* __restrict__ O0,  const float* __restrict__ PRb,
    const float* __restrict__ PIb, const float* __restrict__ MRb,
    const float* __restrict__ MIb, const float* __restrict__ QPR,
    const float* __restrict__ QPI, const float* __restrict__ QMR,
    const float* __restrict__ QMI, const float* __restrict__ WEO,
    const float* __restrict__ wD1, const float* __restrict__ wD2,
    const float* __restrict__ P0,  const float* __restrict__ b0,
    const float* __restrict__ P1,  const float* __restrict__ P2,
    float* __restrict__ CAT1, float* __restrict__ NEWE)
{
    const int e = blockIdx.x;
    const int t = threadIdx.x;
    __shared__ float sD1[9], sD2[25], op1[48], om1[48], op2[16], om2[16],
                     sc0[64], v1[96], v2[80];

    if (t < 9)  sD1[t] = wD1[(size_t)e * 9 + t];
    if (t < 25) sD2[t] = wD2[(size_t)e * 25 + t];
    if (t < 48) {
        op1[t] = PRb[(size_t)e * 48 + t] - MIb[(size_t)e * 48 + t];
        om1[t] = PIb[(size_t)e * 48 + t] + MRb[(size_t)e * 48 + t];
    }
    if (t < 16) {
        op2[t] = QPR[(size_t)e * 16 + t] - QMI[(size_t)e * 16 + t];
        om2[t] = QPI[(size_t)e * 16 + t] + QMR[(size_t)e * 16 + t];
    }
    sc0[t] = silu_f(O0[(size_t)e * 160 + t]);   // t in [0,64)
    __syncthreads();

    if (t < 32) {            // z1 (j: om1, o0_1, op1) -> D1^T -> * gate
        const float gate = sigm_f(O0[(size_t)e * 160 + 64 + t]);
        const float j0 = om1[t];
        const float j1 = O0[(size_t)e * 160 + 112 + t];
        const float j2 = op1[t];
#pragma unroll
        for (int i = 0; i < 3; ++i)
            v1[t * 3 + i] = (sD1[i] * j0 + sD1[3 + i] * j1 + sD1[6 + i] * j2) * gate;
    } else if (t < 48) {     // z2 (om2, om1_hi, o0_2, op1_hi, op2) -> D2^T -> * gate
        const int c = t - 32;
        const float gate = sigm_f(O0[(size_t)e * 160 + 96 + c]);
        const float j0 = om2[c];
        const float j1 = om1[32 + c];
        const float j2 = O0[(size_t)e * 160 + 144 + c];
        const float j3 = op1[32 + c];
        const float j4 = op2[c];
#pragma unroll
        for (int i = 0; i < 5; ++i)
            v2[c * 5 + i] = (sD2[i] * j0 + sD2[5 + i] * j1 + sD2[10 + i] * j2 +
                             sD2[15 + i] * j3 + sD2[20 + i] * j4) * gate;
    }
    __syncthreads();

    // P0 matvec (64x64): thread t = output d
    {
        float s = b0[t];
        for (int c = 0; c < 64; ++c) s += sc0[c] * P0[c * 64 + t];
        CAT1[(size_t)e * 192 + 128 + t] = s;                  // "scalars"
        const float w = WEO[(size_t)e * 112 + t];
        NEWE[(size_t)e * 240 + t] = s * w;
    }
    if (t < 32) {            // P1 channel mix per m
        const float wd = WEO[(size_t)e * 112 + 64 + t];
#pragma unroll
        for (int m = 0; m < 3; ++m) {
            float a = 0.f;
            for (int c = 0; c < 32; ++c) a += v1[c * 3 + m] * P1[c * 32 + t];
            NEWE[(size_t)e * 240 + 64 + t * 3 + m] = a * wd;
        }
    } else if (t < 48) {     // P2 channel mix per m
        const int d = t - 32;
        const float wd = WEO[(size_t)e * 112 + 96 + d];
#pragma unroll
        for (int m = 0; m < 5; ++m) {
            float a = 0.f;
            for (int c = 0; c < 16; ++c) a += v2[c * 5 + m] * P2[c * 16 + d];
            NEWE[(size_t)e * 240 + 160 + d * 5 + m] = a * wd;
        }
    }
}

// ---------------------------------------------------------------------------
// Final: blend edge features, apply one-hot T scaling, write latents output.
// ---------------------------------------------------------------------------
__global__ __launch_bounds__(256) void final_kernel(
    const float* __restrict__ NEWE, const float* __restrict__ edgef,
    const float* __restrict__ TBUF, const float* __restrict__ NL,
    const float* __restrict__ cutoff, const float* __restrict__ latents,
    const int* __restrict__ aedges,
    float* __restrict__ out_edge, float* __restrict__ out_lat)
{
    const int e = blockIdx.x;
    const int t = threadIdx.x;
    const float C_OLD = 0.8944271909999159f;    // rsqrt(1.25)
    const float C_NEW = 0.4472135954999579f;    // 0.5 * C_OLD
    const float S     = 0.08838834764831845f;   // 1/sqrt(128)

    if (t < 240) {
        const float eo = C_OLD * edgef[(size_t)e * 240 + t] +
                         C_NEW * NEWE[(size_t)e * 240 + t];
        int ti;
        if (t < 64)       ti = t;
        else if (t < 160) ti = 64 + (t - 64) / 3;
        else              ti = 96 + (t - 160) / 5;
        const float tv = TBUF[(size_t)e * 112 + ti] * S;
        out_edge[(size_t)e * 240 + t] = eo * (1.f + tv);
    }
    if (t < 128) {
        const int ae = aedges[e];
        const float c  = cutoff[ae];
        const float nl = NL[(size_t)e * 128 + t] * c;
        out_lat[(size_t)ae * 128 + t] =
            C_NEW * nl + C_OLD * latents[(size_t)ae * 128 + t];
    }
}

// ---------------------------------------------------------------------------
extern "C" void kernel_launch(void* const* d_in, const int* in_sizes, int n_in,
                              void* d_out, int out_size, void* d_ws, size_t ws_size,
                              hipStream_t stream)
{
    const size_t E = E_CNT;
    const float* latents = (const float*)d_in[0];
    const float* nodef   = (const float*)d_in[1];
    const float* edgef   = (const float*)d_in[3];
    const float* cutoff  = (const float*)d_in[5];
    const float* eoh     = (const float*)d_in[6];
    const float* wD1     = (const float*)d_in[7];
    const float* wD2     = (const float*)d_in[8];
    const float* moleg   = (const float*)d_in[9];
    const int*   eidx    = (const int*)d_in[10];
    const int*   aedges  = (const int*)d_in[11];
    const float* Wg   = (const float*)d_in[12];
    const float* W0   = (const float*)d_in[13];
    const float* W1r  = (const float*)d_in[14];
    const float* W1i  = (const float*)d_in[15];
    const float* W2r  = (const float*)d_in[16];
    const float* W2i  = (const float*)d_in[17];
    const float* P0   = (const float*)d_in[18];
    const float* b0   = (const float*)d_in[19];
    const float* P1   = (const float*)d_in[20];
    const float* P2   = (const float*)d_in[21];
    const float* We   = (const float*)d_in[22];
    const float* be   = (const float*)d_in[23];
    const float* ln_g = (const float*)d_in[24];
    const float* ln_b = (const float*)d_in[25];
    const float* M1W0 = (const float*)d_in[26];
    const float* M1b0 = (const float*)d_in[27];
    const float* M1W1 = (const float*)d_in[28];
    const float* M1b1 = (const float*)d_in[29];
    const float* M1W2 = (const float*)d_in[30];
    const float* M1b2 = (const float*)d_in[31];
    const float* M2W0 = (const float*)d_in[32];
    const float* M2b0 = (const float*)d_in[33];
    const float* M2W1 = (const float*)d_in[34];
    const float* M2b1 = (const float*)d_in[35];
    const float* M2W2 = (const float*)d_in[36];
    const float* M2b2 = (const float*)d_in[37];
    const float* T0   = (const float*)d_in[38];
    const float* T1   = (const float*)d_in[39];
    const float* T2   = (const float*)d_in[40];

    float* out      = (float*)d_out;
    float* out_lat  = out + E * 240;
    float* outD1    = out + E * 368;
    float* outD2    = out + E * 377;

    // workspace layout
    float* ws = (float*)d_ws;
    size_t off = 0;
    auto alloc = [&](size_t n) { float* p = ws + off; off += n; return p; };
    float* F0   = alloc(E * 336);
    float* F1   = alloc(E * 288);
    float* F2   = alloc(E * 96);
    float* GF   = alloc(E * 16);
    float* CAT1 = alloc(E * 192);
    float* O0b  = alloc(E * 160);
    float* PRb  = alloc(E * 48);
    float* PIb  = alloc(E * 48);
    float* MRb  = alloc(E * 48);
    float* MIb  = alloc(E * 48);
    float* QPR  = alloc(E * 16);
    float* QPI  = alloc(E * 16);
    float* QMR  = alloc(E * 16);
    float* QMI  = alloc(E * 16);
    float* WEO  = alloc(E * 112);
    float* TBUF = alloc(E * 112);
    float* NEWE = alloc(E * 240);
    // MLP buffers alias the (dead after the MoE stage) F0/F1 region
    float* Ha = F0;
    float* Hb = F0 + E * 128;
    float* NL = F0 + E * 256;
    (void)in_sizes; (void)n_in; (void)out_size; (void)ws_size;

    auto gemm = [&](const float* A1, int lda1, int K1,
                    const float* A2, int lda2, int K2, const int* aidx,
                    const float* W, int wks, int wrs, int wcs,
                    int NK, const float* gf, const float* bias, int act,
                    float* outp, int ostride, int Ocols) {
        dim3 grid((unsigned)(E / 32), (unsigned)(Ocols / 16));
        moe_gemm_kernel<<<grid, 32, 0, stream>>>(A1, lda1, K1, A2, lda2, K2, aidx,
                                                 W, wks, wrs, wcs, NK, gf,
                                                 bias, act, outp, ostride);
    };

    // 1) prep
    prep_kernel<<<(unsigned)E, 128, 0, stream>>>(latents, nodef, edgef, wD1, wD2,
        moleg, Wg, ln_g, ln_b, eidx, aedges, F0, F1, F2, GF, CAT1, outD1, outD2);

    // 2) MoE GEMMs (9 experts contracted with gf in-accumulator)
    gemm(F0,       336, 336, nullptr, 0, 0, nullptr, W0,  336 * 160, 160, 1, 9, GF, nullptr, 0, O0b, 160, 160);
    gemm(F1,       288, 144, nullptr, 0, 0, nullptr, W1r, 144 * 48,  48,  1, 9, GF, nullptr, 0, PRb, 48, 48);
    gemm(F1,       288, 144, nullptr, 0, 0, nullptr, W1i, 144 * 48,  48,  1, 9, GF, nullptr, 0, PIb, 48, 48);
    gemm(F1 + 144, 288, 144, nullptr, 0, 0, nullptr, W1r, 144 * 48,  48,  1, 9, GF, nullptr, 0, MRb, 48, 48);
    gemm(F1 + 144, 288, 144, nullptr, 0, 0, nullptr, W1i, 144 * 48,  48,  1, 9, GF, nullptr, 0, MIb, 48, 48);
    gemm(F2,       96,  48,  nullptr, 0, 0, nullptr, W2r, 48 * 16,   16,  1, 9, GF, nullptr, 0, QPR, 16, 16);
    gemm(F2,       96,  48,  nullptr, 0, 0, nullptr, W2i, 48 * 16,   16,  1, 9, GF, nullptr, 0, QPI, 16, 16);
    gemm(F2 + 48,  96,  48,  nullptr, 0, 0, nullptr, W2r, 48 * 16,   16,  1, 9, GF, nullptr, 0, QMR, 16, 16);
    gemm(F2 + 48,  96,  48,  nullptr, 0, 0, nullptr, W2i, 48 * 16,   16,  1, 9, GF, nullptr, 0, QMI, 16, 16);

    // 3) latent scale (row-gathered A) and one-hot T projection (B transposed)
    gemm(latents, 128, 128, nullptr, 0, 0, aedges, We, 0, 112, 1, 1, nullptr, be, 0, WEO, 112, 112);
    gemm(eoh, 128, 128, nullptr, 0, 0, nullptr, T0, 0, 1, 128, 1, nullptr, nullptr, 0, TBUF,      112, 64);
    gemm(eoh, 128, 128, nullptr, 0, 0, nullptr, T1, 0, 1, 128, 1, nullptr, nullptr, 0, TBUF + 64, 112, 32);
    gemm(eoh, 128, 128, nullptr, 0, 0, nullptr, T2, 0, 1, 128, 1, nullptr, nullptr, 0, TBUF + 96, 112, 16);

    // 4) middle epilogue (rotations back, gates, P0/P1/P2, We scaling)
    middle_kernel<<<(unsigned)E, 64, 0, stream>>>(O0b, PRb, PIb, MRb, MIb,
        QPR, QPI, QMR, QMI, WEO, wD1, wD2, P0, b0, P1, P2, CAT1, NEWE);

    // 5) MLP1: [ln | scalars](192) -> 128 -> 128 -> 128
    gemm(CAT1, 192, 192, nullptr, 0, 0, nullptr, M1W0, 0, 128, 1, 1, nullptr, M1b0, 1, Ha, 128, 128);
    gemm(Ha,   128, 128, nullptr, 0, 0, nullptr, M1W1, 0, 128, 1, 1, nullptr, M1b1, 1, Hb, 128, 128);
    gemm(Hb,   128, 128, nullptr, 0, 0, nullptr, M1W2, 0, 128, 1, 1, nullptr, M1b2, 0, NL, 128, 128);
    // 6) MLP2: [nl | edge_one_hot](256) -> 128 -> 128 -> 128
    gemm(NL, 128, 128, eoh, 128, 128, nullptr, M2W0, 0, 128, 1, 1, nullptr, M2b0, 1, Ha, 128, 128);
    gemm(Ha, 128, 128, nullptr, 0, 0,  nullptr, M2W1, 0, 128, 1, 1, nullptr, M2b1, 1, Hb, 128, 128);
    gemm(Hb, 128, 128, nullptr, 0, 0,  nullptr, M2W2, 0, 128, 1, 1, nullptr, M2b2, 0, NL, 128, 128);

    // 7) final blend + outputs
    final_kernel<<<(unsigned)E, 256, 0, stream>>>(NEWE, edgef, TBUF, NL, cutoff,
        latents, aedges, out, out_lat);
}